// ImprovedGNN_62500364091583
// MI455X (gfx1250) — compile-verified
//
#include <hip/hip_runtime.h>
#include <cmath>

// GCN forward for MI455X (gfx1250): wave32, WMMA f32 16x16x4 for all dense GEMMs.
typedef float v2f __attribute__((ext_vector_type(2)));
typedef float v8f __attribute__((ext_vector_type(8)));

#define NN 50000     // nodes
#define NF 128       // in features / hidden
#define NHALF 64
#define NC 40
#define BN_EPS 1e-5f

// ---------------------------------------------------------------- init / degree
__global__ void k_init(float* __restrict__ deg, float* __restrict__ stats) {
    int i = blockIdx.x * blockDim.x + threadIdx.x;
    if (i < NN) deg[i] = 1.0f;          // self-loop contributes 1 to degree
    if (i < 512) stats[i] = 0.0f;       // 2 layers x (sum[128], sumsq[128])
}

__global__ void k_deg_count(const int* __restrict__ dst, float* __restrict__ deg, int n) {
    int i = blockIdx.x * blockDim.x + threadIdx.x;
    if (i < n) atomicAdd(&deg[dst[i]], 1.0f);
}

__global__ void k_rsqrt(float* __restrict__ deg) {
    int i = blockIdx.x * blockDim.x + threadIdx.x;
    if (i < NN) deg[i] = rsqrtf(deg[i]);
}

// ---------------------------------------------------------------- WMMA GEMM
// C[M,N] = A[M,128] @ Wt[N,128]^T  (+bias, +relu).  M tiles on grid.x (16 rows/block).
// Per ISA 7.12.2 (32-bit 16x4 A): lane = m (0-15) | half<<4 ; vgpr0 = K(0|2), vgpr1 = K(1|3).
// B 4x16 mirrors A; C/D v8f: M = r + 8*half, N = lane&15.
template <int N, bool BIAS, bool RELU>
__global__ void k_gemm128(const float* __restrict__ A, const float* __restrict__ Wt,
                          const float* __restrict__ bias, float* __restrict__ C) {
    constexpr int K = 128;
    __shared__ float sA[16 * K];                 // 8 KB staged X tile
    const int m0 = blockIdx.x * 16;

    for (int i = threadIdx.x; i < 16 * (K / 4); i += blockDim.x) {
        int r = i >> 5, c4 = i & 31;             // K/4 == 32 float4 per row
        *(float4*)(sA + r * K + c4 * 4) =
            *(const float4*)(A + (size_t)(m0 + r) * K + c4 * 4);
    }
    __syncthreads();

    const int wave = threadIdx.x >> 5;
    const int lane = threadIdx.x & 31;
    if (wave >= N / 16) return;
    const int n0 = wave * 16;
    const int row = lane & 15, half = lane >> 4;

    const float* a = sA + row * K + 2 * half;                     // LDS (ds_load_b64)
    const float* b = Wt + (size_t)(n0 + row) * K + 2 * half;      // L2-resident weights

    v8f acc0 = {}, acc1 = {};                     // two chains for ILP
#pragma unroll
    for (int k = 0; k < K; k += 8) {
        v2f av0 = *(const v2f*)(a + k);
        v2f bv0 = *(const v2f*)(b + k);
        acc0 = __builtin_amdgcn_wmma_f32_16x16x4_f32(false, av0, false, bv0,
                                                     (short)0, acc0, false, false);
        v2f av1 = *(const v2f*)(a + k + 4);
        v2f bv1 = *(const v2f*)(b + k + 4);
        acc1 = __builtin_amdgcn_wmma_f32_16x16x4_f32(false, av1, false, bv1,
                                                     (short)0, acc1, false, false);
    }

    float bv = BIAS ? bias[n0 + row] : 0.0f;
#pragma unroll
    for (int r = 0; r < 8; ++r) {
        float v = acc0[r] + acc1[r] + bv;
        if (RELU) v = fmaxf(v, 0.0f);
        C[(size_t)(m0 + r + 8 * half) * N + n0 + row] = v;
    }
}

// ---------------------------------------------------------------- aggregation
// out[i,f] = bias[f] + h[i,f] * dis[i]^2      (self-loop term + bias)
__global__ void k_agg_init(const float* __restrict__ h, const float* __restrict__ dis,
                           const float* __restrict__ bias, float* __restrict__ out) {
    size_t i = (size_t)blockIdx.x * blockDim.x + threadIdx.x;
    if (i >= (size_t)NN * NF) return;
    int f = (int)(i & (NF - 1));
    int r = (int)(i >> 7);
    float w = dis[r]; w *= w;
    out[i] = bias[f] + h[i] * w;
}

// one wave32 per edge: lane handles 4 feats (float4 load + 4 f32 global atomics)
__global__ void k_edge_scatter(const float* __restrict__ h, const int* __restrict__ src,
                               const int* __restrict__ dst, const float* __restrict__ dis,
                               float* __restrict__ out, int nedges) {
    int wave = (blockIdx.x * blockDim.x + threadIdx.x) >> 5;
    int lane = threadIdx.x & 31;
    if (wave >= nedges) return;
    int s = src[wave], d = dst[wave];
    float norm = dis[s] * dis[d];
    float4 v = *(const float4*)(h + (size_t)s * NF + lane * 4);
    float* o = out + (size_t)d * NF + lane * 4;
    atomicAdd(o + 0, v.x * norm);
    atomicAdd(o + 1, v.y * norm);
    atomicAdd(o + 2, v.z * norm);
    atomicAdd(o + 3, v.w * norm);
}

// ---------------------------------------------------------------- batch norm
__global__ void k_bn_stats(const float* __restrict__ x, float* __restrict__ stats) {
    __shared__ float s_sum[256], s_sq[256];
    int f = threadIdx.x & 127;
    int g = threadIdx.x >> 7;                    // two row streams per feature
    float acc = 0.f, acc2 = 0.f;
    for (int r = blockIdx.x * 2 + g; r < NN; r += gridDim.x * 2) {
        float v = x[(size_t)r * NF + f];
        acc += v; acc2 += v * v;
    }
    s_sum[threadIdx.x] = acc; s_sq[threadIdx.x] = acc2;
    __syncthreads();
    if (threadIdx.x < 128) {
        atomicAdd(&stats[f],       s_sum[threadIdx.x] + s_sum[threadIdx.x + 128]);
        atomicAdd(&stats[128 + f], s_sq[threadIdx.x]  + s_sq[threadIdx.x + 128]);
    }
}

__global__ void k_bn_relu(float* __restrict__ x, const float* __restrict__ stats,
                          const float* __restrict__ gamma, const float* __restrict__ beta) {
    size_t i = (size_t)blockIdx.x * blockDim.x + threadIdx.x;
    if (i >= (size_t)NN * NF) return;
    int f = (int)(i & (NF - 1));
    const float inv_n = 1.0f / (float)NN;
    float mean = stats[f] * inv_n;
    float var  = stats[128 + f] * inv_n - mean * mean;   // biased var, like jnp.var
    float scale = gamma[f] * rsqrtf(var + BN_EPS);
    float shift = beta[f] - mean * scale;
    float v = x[i] * scale + shift;
    x[i] = v > 0.0f ? v : 0.0f;
}

// ---------------------------------------------------------------- FC2 + log_softmax
// one wave32 per node: lanes 0-31 own classes 0-31; lanes 0-7 also own 32-39.
__global__ void k_fc2_logsoftmax(const float* __restrict__ h, const float* __restrict__ W,
                                 const float* __restrict__ b, float* __restrict__ out) {
    int wave = (blockIdx.x * blockDim.x + threadIdx.x) >> 5;
    int lane = threadIdx.x & 31;
    if (wave >= NN) return;
    const float* hr = h + (size_t)wave * NHALF;
    int c1 = 32 + (lane & 7);
    float acc0 = b[lane];
    float acc1 = b[c1];
#pragma unroll 8
    for (int k = 0; k < NHALF; ++k) {
        float hv = hr[k];                         // broadcast load
        acc0 += hv * W[lane * NHALF + k];
        acc1 += hv * W[c1 * NHALF + k];
    }
    bool has2 = lane < 8;
    float m = fmaxf(acc0, has2 ? acc1 : -INFINITY);
#pragma unroll
    for (int o = 16; o >= 1; o >>= 1) m = fmaxf(m, __shfl_xor(m, o, 32));
    float e = __expf(acc0 - m) + (has2 ? __expf(acc1 - m) : 0.0f);
#pragma unroll
    for (int o = 16; o >= 1; o >>= 1) e += __shfl_xor(e, o, 32);
    float lse = m + __logf(e);
    float* orow = out + (size_t)wave * NC;
    orow[lane] = acc0 - lse;
    if (has2) orow[32 + lane] = acc1 - lse;
}

// ---------------------------------------------------------------- launcher
extern "C" void kernel_launch(void* const* d_in, const int* in_sizes, int n_in,
                              void* d_out, int out_size, void* d_ws, size_t ws_size,
                              hipStream_t stream) {
    const float* x   = (const float*)d_in[0];
    const int*   ei  = (const int*)d_in[1];      // [2,E] int32 per harness convention
    const float* W1  = (const float*)d_in[2];
    const float* b1  = (const float*)d_in[3];
    const float* W2  = (const float*)d_in[4];
    const float* b2  = (const float*)d_in[5];
    const float* g1  = (const float*)d_in[6];
    const float* be1 = (const float*)d_in[7];
    const float* g2  = (const float*)d_in[8];
    const float* be2 = (const float*)d_in[9];
    const float* Wf1 = (const float*)d_in[10];
    const float* bf1 = (const float*)d_in[11];
    const float* Wf2 = (const float*)d_in[12];
    const float* bf2 = (const float*)d_in[13];
    float* out = (float*)d_out;

    const int E = in_sizes[1] / 2;
    const int* src = ei;
    const int* dst = ei + E;

    float* w     = (float*)d_ws;
    float* deg   = w;                            // [50000]  (becomes deg_isqrt)
    float* stats = w + 50048;                    // [512] : L1 sum/sq, L2 sum/sq
    float* tmp1  = w + 51200;                    // [50000*128]
    float* tmp2  = tmp1 + (size_t)NN * NF;       // [50000*128]
    float* st1 = stats, *st2 = stats + 256;

    const int EW_GRID   = (int)(((size_t)NN * NF + 255) / 256);   // 25000
    const int MTILES    = NN / 16;                                // 3125
    const int SCAT_GRID = (E * 32 + 255) / 256;                   // 8 waves/block

    // degree / normalization
    k_init<<<(NN + 255) / 256, 256, 0, stream>>>(deg, stats);
    k_deg_count<<<(E + 255) / 256, 256, 0, stream>>>(dst, deg, E);
    k_rsqrt<<<(NN + 255) / 256, 256, 0, stream>>>(deg);

    // layer 1: GCNConv + BN + ReLU
    k_gemm128<NF, false, false><<<MTILES, 256, 0, stream>>>(x, W1, nullptr, tmp1);
    k_agg_init<<<EW_GRID, 256, 0, stream>>>(tmp1, deg, b1, tmp2);
    k_edge_scatter<<<SCAT_GRID, 256, 0, stream>>>(tmp1, src, dst, deg, tmp2, E);
    k_bn_stats<<<512, 256, 0, stream>>>(tmp2, st1);
    k_bn_relu<<<EW_GRID, 256, 0, stream>>>(tmp2, st1, g1, be1);

    // layer 2: GCNConv + BN + ReLU
    k_gemm128<NF, false, false><<<MTILES, 256, 0, stream>>>(tmp2, W2, nullptr, tmp1);
    k_agg_init<<<EW_GRID, 256, 0, stream>>>(tmp1, deg, b2, tmp2);
    k_edge_scatter<<<SCAT_GRID, 256, 0, stream>>>(tmp1, src, dst, deg, tmp2, E);
    k_bn_stats<<<512, 256, 0, stream>>>(tmp2, st2);
    k_bn_relu<<<EW_GRID, 256, 0, stream>>>(tmp2, st2, g2, be2);

    // head: FC1 (WMMA, fused bias+relu) then FC2 + log_softmax
    k_gemm128<NHALF, true, true><<<MTILES, 128, 0, stream>>>(tmp2, Wf1, bf1, tmp1);
    k_fc2_logsoftmax<<<(NN * 32 + 255) / 256, 256, 0, stream>>>(tmp1, Wf2, bf2, out);
}